// GraphAttentionLayer_5205500363117
// MI455X (gfx1250) — compile-verified
//
#include <hip/hip_runtime.h>
#include <math.h>

#define NN      10000
#define EE      320000
#define IN_DIM  256
#define OUT_DIM 64
#define ALPHA   0.2f

typedef __attribute__((ext_vector_type(2))) float v2f;
typedef __attribute__((ext_vector_type(8))) float v8f;

// Monotonic float<->uint encoding so atomicMax(uint) == float max (handles negatives).
__device__ __forceinline__ unsigned f2ord(float f) {
  unsigned u = __float_as_uint(f);
  return (u & 0x80000000u) ? ~u : (u | 0x80000000u);
}
__device__ __forceinline__ float ord2f(unsigned o) {
  return (o & 0x80000000u) ? __uint_as_float(o & 0x7fffffffu) : __uint_as_float(~o);
}

// ---------------------------------------------------------------- init
__global__ void init_kernel(float* __restrict__ out, float* __restrict__ rowsum,
                            unsigned* __restrict__ ordmax, float* __restrict__ colsum) {
  int i = blockIdx.x * blockDim.x + threadIdx.x;
  if (i < NN * OUT_DIM) out[i] = 0.0f;
  if (i < NN) { rowsum[i] = 0.0f; ordmax[i] = 0u; }  // ord==0 < ord(-inf): zero-degree sentinel
  if (i < OUT_DIM) colsum[i] = 0.0f;
}

// ---------------------------------------------------------------- Wh = X@W (WMMA fp32) + fused dots
#define XS_STRIDE  260   // 256 + 4 pad: avoids 16-way LDS bank conflict on column reads
#define WHS_STRIDE 65

__launch_bounds__(128)
__global__ void wh_wmma_kernel(const float* __restrict__ X, const float* __restrict__ W,
                               const float* __restrict__ a,
                               float* __restrict__ Wh, float* __restrict__ s_src,
                               float* __restrict__ s_dst, float* __restrict__ colsum) {
  __shared__ float Xs[16 * XS_STRIDE];
  __shared__ float Whs[16 * WHS_STRIDE];
  __shared__ float a_sm[2 * OUT_DIM];

  const int tid = threadIdx.x;
  const int m0  = blockIdx.x * 16;

  // Stage 16x256 X tile into LDS (flat coalesced copy).
  for (int i = tid; i < 16 * IN_DIM; i += 128) {
    int r = i >> 8;       // / 256
    int k = i & 255;
    Xs[r * XS_STRIDE + k] = X[(m0 + r) * IN_DIM + k];
  }
  if (tid < 2 * OUT_DIM) a_sm[tid] = a[tid];
  __syncthreads();

  const int wave = tid >> 5;
  const int lane = tid & 31;
  const int half = lane >> 4;      // 0: K offset 0,1   1: K offset 2,3
  const int l16  = lane & 15;
  const int ncol = wave * 16 + l16;  // this wave's 16-column tile, lane's column

  v8f c = {};
  for (int k0 = 0; k0 < IN_DIM; k0 += 4) {
    const int ka = k0 + half * 2;
    v2f av, bv;
    // A 16x4 frag: lane row = l16; VGPR0=K=ka, VGPR1=K=ka+1
    const float* xr = &Xs[l16 * XS_STRIDE + ka];
    av.x = xr[0];
    av.y = xr[1];
    // B 4x16 frag (mirror layout): lane col = ncol; VGPR0=K=ka row, VGPR1=K=ka+1 row
    const float* wr = &W[ka * OUT_DIM + ncol];
    bv.x = wr[0];
    bv.y = wr[OUT_DIM];
    c = __builtin_amdgcn_wmma_f32_16x16x4_f32(false, av, false, bv, (short)0, c,
                                              false, false);
  }

  // C/D layout: VGPR r, lane -> element (M = r + 8*half, N = l16-in-tile)
#pragma unroll
  for (int r = 0; r < 8; ++r) {
    const int m = r + 8 * half;
    const float v = c[r];
    Wh[(m0 + m) * OUT_DIM + ncol] = v;
    Whs[m * WHS_STRIDE + ncol] = v;
  }
  __syncthreads();

  // Fused per-row attention dots: s_src = Wh . a[:64], s_dst = Wh . a[64:]
  if (tid < 16) {
    float ss = 0.0f, sd = 0.0f;
#pragma unroll 8
    for (int j = 0; j < OUT_DIM; ++j) {
      const float v = Whs[tid * WHS_STRIDE + j];
      ss += v * a_sm[j];
      sd += v * a_sm[OUT_DIM + j];
    }
    s_src[m0 + tid] = ss;
    s_dst[m0 + tid] = sd;
  }
  // Column partial sums (for zero-out-degree softmax fallback = column mean).
  if (tid < OUT_DIM) {
    float p = 0.0f;
#pragma unroll
    for (int r = 0; r < 16; ++r) p += Whs[r * WHS_STRIDE + tid];
    atomicAdd(&colsum[tid], p);
  }
}

// ---------------------------------------------------------------- edge passes
__global__ void edge_score_kernel(const int* __restrict__ src, const int* __restrict__ dst,
                                  const float* __restrict__ s_src,
                                  const float* __restrict__ s_dst,
                                  float* __restrict__ escore, unsigned* __restrict__ ordmax) {
  int e = blockIdx.x * blockDim.x + threadIdx.x;
  if (e >= EE) return;
  const int s = src[e];
  float v = s_src[s] + s_dst[dst[e]];
  v = (v > 0.0f) ? v : ALPHA * v;  // leaky relu
  escore[e] = v;
  atomicMax(&ordmax[s], f2ord(v));  // row-wise (segmented) max
}

__global__ void edge_expsum_kernel(const int* __restrict__ src,
                                   const unsigned* __restrict__ ordmax,
                                   float* __restrict__ escore,  // in: score, out: exp weight
                                   float* __restrict__ rowsum) {
  int e = blockIdx.x * blockDim.x + threadIdx.x;
  if (e >= EE) return;
  const int s = src[e];
  const float w = expf(escore[e] - ord2f(ordmax[s]));
  escore[e] = w;
  atomicAdd(&rowsum[s], w);
}

// One wave32 per edge: lanes cover channels {lane, lane+32}.
__global__ void edge_agg_kernel(const int* __restrict__ src, const int* __restrict__ dst,
                                const float* __restrict__ ew, const float* __restrict__ rowsum,
                                const float* __restrict__ Wh, float* __restrict__ out) {
  const int wid  = (blockIdx.x * blockDim.x + threadIdx.x) >> 5;
  const int lane = threadIdx.x & 31;
  if (wid >= EE) return;
  const int s = src[wid];
  const int d = dst[wid];
  const float coef = ew[wid] / rowsum[s];
  const float v0 = Wh[d * OUT_DIM + lane];
  const float v1 = Wh[d * OUT_DIM + lane + 32];
  atomicAdd(&out[s * OUT_DIM + lane],      coef * v0);
  atomicAdd(&out[s * OUT_DIM + lane + 32], coef * v1);
}

// Zero-out-degree rows: softmax over all-NEG_FILL row is uniform -> column mean of Wh.
__global__ void finalize_kernel(const unsigned* __restrict__ ordmax,
                                const float* __restrict__ colsum, float* __restrict__ out) {
  int i = blockIdx.x * blockDim.x + threadIdx.x;
  if (i >= NN * OUT_DIM) return;
  const int n = i >> 6;
  const int j = i & 63;
  if (ordmax[n] == 0u) out[i] = colsum[j] * (1.0f / (float)NN);
}

// ---------------------------------------------------------------- launch
extern "C" void kernel_launch(void* const* d_in, const int* in_sizes, int n_in,
                              void* d_out, int out_size, void* d_ws, size_t ws_size,
                              hipStream_t stream) {
  const float* X     = (const float*)d_in[0];
  const int*   edges = (const int*)d_in[1];   // (2, E): [src | dst]
  const float* W     = (const float*)d_in[2];
  const float* a     = (const float*)d_in[3];
  float* out = (float*)d_out;
  const int* src = edges;
  const int* dst = edges + EE;

  char* ws = (char*)d_ws;
  float*    Wh     = (float*)(ws);                 // 10000*64*4  = 2,560,000 B
  float*    s_src  = (float*)(ws + 2560000);       // 40,000 B
  float*    s_dst  = (float*)(ws + 2600000);       // 40,000 B
  float*    escore = (float*)(ws + 2640000);       // E*4 = 1,280,000 B (reused as exp weights)
  float*    rowsum = (float*)(ws + 3920000);       // 40,000 B
  unsigned* ordmax = (unsigned*)(ws + 3960000);    // 40,000 B
  float*    colsum = (float*)(ws + 4000000);       // 256 B   -> ~4.0 MB total

  init_kernel<<<(NN * OUT_DIM + 255) / 256, 256, 0, stream>>>(out, rowsum, ordmax, colsum);
  wh_wmma_kernel<<<NN / 16, 128, 0, stream>>>(X, W, a, Wh, s_src, s_dst, colsum);
  edge_score_kernel<<<(EE + 255) / 256, 256, 0, stream>>>(src, dst, s_src, s_dst, escore, ordmax);
  edge_expsum_kernel<<<(EE + 255) / 256, 256, 0, stream>>>(src, ordmax, escore, rowsum);
  edge_agg_kernel<<<(EE * 32 + 255) / 256, 256, 0, stream>>>(src, dst, escore, rowsum, Wh, out);
  finalize_kernel<<<(NN * OUT_DIM + 255) / 256, 256, 0, stream>>>(ordmax, colsum, out);
}